// TSCoupler_70300024701852
// MI455X (gfx1250) — compile-verified
//
#include <hip/hip_runtime.h>
#include <hip/hip_bf16.h>

typedef float v2f __attribute__((ext_vector_type(2)));
typedef float v8f __attribute__((ext_vector_type(8)));

#define LSIG 8192
#define NCH  21
#define KW   15
#define PI_F 3.14159265358979323846f
#define TWO_PI_F 6.28318530717958647693f

#if __has_builtin(__builtin_amdgcn_global_load_async_to_lds_b32)
#define HAVE_ASYNC_LDS 1
#else
#define HAVE_ASYNC_LDS 0
#endif

typedef __attribute__((address_space(1))) int gint_t;   // global (AS1)
typedef __attribute__((address_space(3))) int lint_t;   // LDS (AS3)

// ---------------- table layout in d_ws (floats) ----------------
// C128[16384] @ 0        cos(2*pi*n*k/128),  n,k in [0,128)
// S128[16384] @ 16384    sin(...)
// C64 [ 4096] @ 32768    cos(2*pi*n*k/64)
// S64 [ 4096] @ 36864    sin(...)
// TWC [ 8192] @ 40960    cos(2*pi*n1*k2/8192), n1 in [0,64), k2 in [0,128)
// TWS [ 8192] @ 49152    sin(...)
// total 57344 floats = 224 KB (L2-resident on MI455X's 192MB L2)

__global__ void tsc_fill_tables(float* __restrict__ tabs) {
    int idx = blockIdx.x * 256 + threadIdx.x;
    const double TP = 6.283185307179586476925286766559;
    if (idx < 16384) {
        int n = idx >> 7, k = idx & 127;
        double ang = TP * (double)((n * k) & 127) / 128.0;
        tabs[idx]          = (float)cos(ang);
        tabs[16384 + idx]  = (float)sin(ang);
    } else if (idx < 16384 + 4096) {
        int j = idx - 16384;
        int n = j >> 6, k = j & 63;
        double ang = TP * (double)((n * k) & 63) / 64.0;
        tabs[32768 + j] = (float)cos(ang);
        tabs[36864 + j] = (float)sin(ang);
    } else if (idx < 16384 + 4096 + 8192) {
        int j = idx - 20480;
        int n1 = j >> 7, k2 = j & 127;
        double ang = TP * (double)((n1 * k2) & 8191) / 8192.0;
        tabs[40960 + j] = (float)cos(ang);
        tabs[49152 + j] = (float)sin(ang);
    }
}

// ---------------------------------------------------------------
// Complex GEMM: OUT[64 x 128] = A[64 x K] * B[K x 128]  (+ optional scale)
// A fragments from (Are, Aim*asign), B fragments from (Bre, Bim*bsign).
// Aim == nullptr means A is purely real (forward step 1).
// Uses V_WMMA_F32_16X16X4_F32: A 16x4 (2 VGPR), B 4x16 (2 VGPR), C/D 8 VGPR.
// Per ISA 7.12.2:  A: lane m=l%16, v0=K(k0+2h), v1=K(k0+2h+1), h=lane/16
//                  C/D: vgpr i -> row m0+i+8h, col n0+l%16
// ---------------------------------------------------------------
__device__ __forceinline__ void gemm64x128(
    const float* __restrict__ Are, const float* __restrict__ Aim, int lda, float asign,
    const float* __restrict__ Bre, const float* __restrict__ Bim, int ldb, float bsign,
    float* __restrict__ Ore, float* __restrict__ Oim, int K, float scale)
{
    const int tid  = threadIdx.x;
    const int w    = tid >> 5;
    const int lane = tid & 31;
    const int hf   = lane >> 4;     // which half of the wave
    const int l16  = lane & 15;
    const int m0    = (w >> 1) * 16;   // 4 row tiles across 8 waves
    const int nbase = (w & 1) * 64;    // 2 column groups

    for (int nt = 0; nt < 4; ++nt) {
        const int n0 = nbase + nt * 16;
        v8f accR = {0.f,0.f,0.f,0.f,0.f,0.f,0.f,0.f};
        v8f accI = {0.f,0.f,0.f,0.f,0.f,0.f,0.f,0.f};
        const int arow = (m0 + l16) * lda;
        for (int k0 = 0; k0 < K; k0 += 4) {
            const int ka = k0 + 2 * hf;
            v2f aR; aR[0] = Are[arow + ka]; aR[1] = Are[arow + ka + 1];
            const int b0 = ka * ldb + n0 + l16;
            const int b1 = b0 + ldb;
            v2f bR; bR[0] = Bre[b0];         bR[1] = Bre[b1];
            v2f bI; bI[0] = Bim[b0] * bsign; bI[1] = Bim[b1] * bsign;
            accR = __builtin_amdgcn_wmma_f32_16x16x4_f32(false, aR, false, bR,
                                                         (short)0, accR, false, false);
            accI = __builtin_amdgcn_wmma_f32_16x16x4_f32(false, aR, false, bI,
                                                         (short)0, accI, false, false);
            if (Aim) {
                v2f aI;  aI[0]  = Aim[arow + ka] * asign; aI[1] = Aim[arow + ka + 1] * asign;
                v2f aIn; aIn[0] = -aI[0];                 aIn[1] = -aI[1];
                accR = __builtin_amdgcn_wmma_f32_16x16x4_f32(false, aIn, false, bI,
                                                             (short)0, accR, false, false);
                accI = __builtin_amdgcn_wmma_f32_16x16x4_f32(false, aI, false, bR,
                                                             (short)0, accI, false, false);
            }
        }
        #pragma unroll
        for (int i = 0; i < 8; ++i) {
            const int row = m0 + i + 8 * hf;
            const int col = n0 + l16;
            Ore[row * 128 + col] = accR[i] * scale;
            Oim[row * 128 + col] = accI[i] * scale;
        }
    }
}

__device__ __forceinline__ float wrap_diff(float d) {
    // torch/jnp-style: mod(d+pi, 2pi) - pi  (floor-mod), with (-pi && d>0) -> pi
    float m = d + PI_F;
    float r = m - TWO_PI_F * floorf(m / TWO_PI_F);
    float dm = r - PI_F;
    if (dm == -PI_F && d > 0.f) dm = PI_F;
    return dm;
}

// ---------------------------------------------------------------
// One block = one (b, c) signal. 256 threads = 8 waves.
// LDS: 4 planes of 8192 f32 (128 KB): ping-pong complex buffers.
// ---------------------------------------------------------------
__global__ void __launch_bounds__(256) tsc_main(
    const float* __restrict__ seasonal, const float* __restrict__ trend,
    const float* __restrict__ alpha_log, const float* __restrict__ phi0,
    const float* __restrict__ beta1_log, const float* __restrict__ beta2_log,
    const float* __restrict__ pc_weight, const float* __restrict__ pc_strength,
    const float* __restrict__ tabs, float* __restrict__ out)
{
    extern __shared__ float sm[];
    float* P0 = sm;
    float* P1 = sm + 8192;
    float* P2 = sm + 16384;
    float* P3 = sm + 24576;
    __shared__ float ssum[256];
    __shared__ float sparam[5 + KW];

    const float* C128 = tabs;
    const float* S128 = tabs + 16384;
    const float* C64t = tabs + 32768;
    const float* S64t = tabs + 36864;
    const float* TWC  = tabs + 40960;
    const float* TWS  = tabs + 49152;

    // tables are shared by all 1344 blocks -> keep them hot (global_prefetch_b8)
    __builtin_prefetch(C128, 0, 3);
    __builtin_prefetch(S128, 0, 3);
    __builtin_prefetch(C64t, 0, 3);
    __builtin_prefetch(TWC,  0, 3);

    const int bc  = blockIdx.x;
    const int c   = bc % NCH;
    const int b   = bc / NCH;
    const int tid = threadIdx.x;

    // stage: D[n1][n2] = x[n1 + 64*n2]  (n = n1 + 64*n2), strided channel gather.
    // Per-lane async copy global->LDS (ASYNCcnt), overlapped with the scalar
    // parameter setup below; seasonal fits in the 192MB L2, so the 21 sibling
    // c-blocks hit the same cachelines.
    {
        const float* gp = seasonal + (size_t)b * LSIG * NCH + c;
        for (int n = tid; n < LSIG; n += 256) {
            const int n1 = n & 63, n2 = n >> 6;
#if HAVE_ASYNC_LDS
            __builtin_amdgcn_global_load_async_to_lds_b32(
                (gint_t*)(gp + (size_t)n * NCH),
                (lint_t*)&P0[n1 * 128 + n2], 0, 0);
#else
            P0[n1 * 128 + n2] = gp[(size_t)n * NCH];
#endif
        }
    }

    if (tid == 0) {
        float mean = 0.f;
        for (int k = 0; k < KW; ++k) mean += pc_weight[c * KW + k];
        mean /= (float)KW;
        sparam[0] = tanhf(pc_strength[0]);                       // gate
        sparam[1] = log1pf(expf(alpha_log[c])) + 1e-6f;          // alpha_c
        sparam[2] = log1pf(expf(beta1_log[0])) + 1e-6f;          // beta1
        sparam[3] = log1pf(expf(beta2_log[0])) + 1e-6f;          // beta2
        sparam[4] = phi0[c];
        for (int k = 0; k < KW; ++k) sparam[5 + k] = pc_weight[c * KW + k] - mean;
    }

#if HAVE_ASYNC_LDS
#if __has_builtin(__builtin_amdgcn_s_wait_asynccnt)
    __builtin_amdgcn_s_wait_asynccnt(0);
#else
    asm volatile("s_wait_asynccnt 0x0" ::: "memory");
#endif
#endif
    __syncthreads();

    // ---- forward FFT (four-step, 8192 = 64 x 128) ----
    // step 1: E = D x F128, D real, F128 = cos - i sin  -> (P2, P3)
    gemm64x128(P0, nullptr, 128, 1.f, C128, S128, 128, -1.f, P2, P3, 128, 1.f);
    __syncthreads();
    // step 2: twiddle w_N^(-n1*k2)
    for (int idx = tid; idx < LSIG; idx += 256) {
        const float ct = TWC[idx], st = TWS[idx];
        const float re = P2[idx], im = P3[idx];
        P2[idx] = re * ct + im * st;
        P3[idx] = im * ct - re * st;
    }
    __syncthreads();
    // step 3: G = F64 x E' -> (P0, P1). G row-major IS the spectrum: k = 128*k1 + k2.
    gemm64x128(C64t, S64t, 64, -1.f, P2, P3, 128, 1.f, P0, P1, 64, 1.f);
    __syncthreads();

    // ---- Hilbert analytic-signal mask (L even): h = {1, 2,...,2, 1, 0,...,0} ----
    for (int k = tid; k < LSIG; k += 256) {
        const float h = (k == 0 || k == 4096) ? 1.f : (k < 4096 ? 2.f : 0.f);
        P0[k] *= h; P1[k] *= h;
    }
    __syncthreads();

    // ---- inverse FFT (conjugate twiddles, 1/N at the end) ----
    gemm64x128(C64t, S64t, 64, 1.f, P0, P1, 128, 1.f, P2, P3, 64, 1.f);
    __syncthreads();
    for (int idx = tid; idx < LSIG; idx += 256) {
        const float ct = TWC[idx], st = TWS[idx];
        const float re = P2[idx], im = P3[idx];
        P2[idx] = re * ct - im * st;
        P3[idx] = im * ct + re * st;
    }
    __syncthreads();
    gemm64x128(P2, P3, 128, 1.f, C128, S128, 128, 1.f, P0, P1, 128, 1.f / 8192.f);
    __syncthreads();

    // ---- instantaneous phase, time-ordered into P2 (z[n1+64*n2] at [n1*128+n2]) ----
    for (int n = tid; n < LSIG; n += 256) {
        const int n1 = n & 63, n2 = n >> 6;
        P2[n] = atan2f(P1[n1 * 128 + n2], P0[n1 * 128 + n2]);
    }
    __syncthreads();

    // ---- unwrap: U[n] = P[0] + sum_{j<=n} wrap(diff) ; block scan (32 elems/thread) ----
    {
        const int base = tid * 32;
        float s = 0.f;
        for (int j = base; j < base + 32; ++j) {
            if (j == 0) continue;
            s += wrap_diff(P2[j] - P2[j - 1]);
        }
        ssum[tid] = s;
        __syncthreads();
        for (int off = 1; off < 256; off <<= 1) {
            const float v   = ssum[tid];
            const float add = (tid >= off) ? ssum[tid - off] : 0.f;
            __syncthreads();
            ssum[tid] = v + add;
            __syncthreads();
        }
        const float p0  = P2[0];
        float run = (tid > 0) ? ssum[tid - 1] : 0.f;
        for (int j = base; j < base + 32; ++j) {
            if (j == 0) { P3[0] = p0; continue; }
            run += wrap_diff(P2[j] - P2[j - 1]);
            P3[j] = p0 + run;
        }
    }
    __syncthreads();

    // ---- depthwise conv (reflect, K=15) + AM envelope + coalesced [B,C,L] store ----
    const float gate = sparam[0], alpha_c = sparam[1];
    const float b1 = sparam[2], b2 = sparam[3], ph0 = sparam[4];
    for (int n = tid; n < LSIG; n += 256) {
        float delta = 0.f;
        #pragma unroll
        for (int k = 0; k < KW; ++k) {
            int i = n + k - (KW / 2);
            i = (i < 0) ? -i : ((i >= LSIG) ? (2 * LSIG - 2 - i) : i);
            delta += sparam[5 + k] * P3[i];
        }
        const float phi = P3[n] + gate * delta + ph0;
        float T = trend[(size_t)n * NCH + c];           // batch 0 only (L2 reuse across b)
        T = fminf(10.f, fmaxf(-10.f, T));
        const float At = alpha_c * (b1 * log1pf(expf(b2 * T)));
        out[((size_t)(b * NCH + c)) * LSIG + n] = At * cosf(phi);
    }
}

extern "C" void kernel_launch(void* const* d_in, const int* in_sizes, int n_in,
                              void* d_out, int out_size, void* d_ws, size_t ws_size,
                              hipStream_t stream) {
    const float* seasonal    = (const float*)d_in[0];
    const float* trend       = (const float*)d_in[1];
    const float* alpha_log   = (const float*)d_in[2];
    const float* phi0        = (const float*)d_in[3];
    const float* beta1_log   = (const float*)d_in[4];
    const float* beta2_log   = (const float*)d_in[5];
    const float* pc_weight   = (const float*)d_in[6];
    const float* pc_strength = (const float*)d_in[7];
    float* tabs = (float*)d_ws;
    float* out  = (float*)d_out;

    // 1) DFT/twiddle tables (224 KB in d_ws, L2-resident afterwards)
    tsc_fill_tables<<<(57344 + 255) / 256, 256, 0, stream>>>(tabs);

    // 2) main pipeline: 64*21 = 1344 blocks, 256 threads, 128 KB dynamic LDS
    const size_t shmem = 4u * 8192u * sizeof(float);
    (void)hipFuncSetAttribute((const void*)tsc_main,
                              hipFuncAttributeMaxDynamicSharedMemorySize, (int)shmem);
    tsc_main<<<64 * NCH, 256, shmem, stream>>>(seasonal, trend, alpha_log, phi0,
                                               beta1_log, beta2_log, pc_weight,
                                               pc_strength, tabs, out);
}